// FeatureNormMagOnline_60825326846429
// MI455X (gfx1250) — compile-verified
//
#include <hip/hip_runtime.h>
#include <cstdint>

// Problem shape (fixed by the reference)
#define B_ 16
#define C_ 2
#define T_ 1000
#define F_ 257
#define ROWF (F_ * 2)          // 514 floats per time-row of one (b,c)
#define NCHUNK 40              // T split into 40 chunks
#define CHUNK 25               // 40 * 25 == 1000

#if defined(__gfx1250__) && __has_builtin(__builtin_amdgcn_tensor_load_to_lds) && \
    __has_builtin(__builtin_amdgcn_s_wait_tensorcnt)
#define USE_TDM 1
#else
#define USE_TDM 0
#endif

typedef unsigned int u32;
typedef u32 u32x4 __attribute__((ext_vector_type(4)));
typedef int  i32x4 __attribute__((ext_vector_type(4)));
typedef int  i32x8 __attribute__((ext_vector_type(8)));

__device__ __forceinline__ float sigmoidf_(float x) {
    return 1.0f / (1.0f + __expf(-x));
}

#if USE_TDM
// Issue one TDM load of an [nrows x ROWF] f32 tile (contiguous rows) into LDS.
// D# layout per CDNA5 ISA §8 (groups 0..3). 2-D tile; groups 2/3 benign.
// This toolchain's builtin takes 6 args:
//   (u32x4 g0, i32x8 g1, i32x4 g2, i32x4 g3, i32x8 gx, i32 cpol)
__device__ __forceinline__ void tdm_load_rows(const float* gsrc, u32 lds_byte_addr,
                                              u32 nrows) {
    const uint64_t ga = (uint64_t)(uintptr_t)gsrc;

    u32x4 g0;
    g0[0] = 1u;                                        // count=1 (valid), user mode
    g0[1] = lds_byte_addr;                             // lds_addr (bytes)
    g0[2] = (u32)(ga & 0xffffffffu);                   // global_addr[31:0]
    g0[3] = (u32)((ga >> 32) & 0x1ffffffu) | (2u << 30); // addr[56:32] | type=2

    const u32 dim0  = ROWF;                            // tensor_dim0 (elements)
    const u32 dim1  = nrows;                           // tensor_dim1
    const u32 tile0 = ROWF;                            // tile_dim0
    const u32 tile1 = nrows;                           // tile_dim1
    const uint64_t stride0 = ROWF;                     // dim0 stride (elements)
    const uint64_t stride1 = (uint64_t)ROWF * nrows;   // dim1 stride (unused for 2D)

    u32 w[8];
    w[0] = (2u << 16);                                 // data_size=2 -> 4 bytes; mask=0
    w[1] = (dim0 & 0xffffu) << 16;                     // tensor_dim0 lo16 @bit48
    w[2] = (dim0 >> 16) | ((dim1 & 0xffffu) << 16);    // dim0 hi16 | dim1 lo16
    w[3] = (dim1 >> 16) | (tile0 << 16);               // dim1 hi16 | tile_dim0
    w[4] = tile1;                                      // tile_dim1 | tile_dim2=0
    w[5] = (u32)(stride0 & 0xffffffffu);               // dim0_stride lo32
    w[6] = (u32)((stride0 >> 32) & 0xffffu) |
           ((u32)(stride1 & 0xffffu) << 16);           // dim0_stride hi16 | dim1_stride lo16
    w[7] = (u32)((stride1 >> 16) & 0xffffffffu);       // dim1_stride [47:16]

    i32x8 g1;
    g1[0] = (int)w[0]; g1[1] = (int)w[1]; g1[2] = (int)w[2]; g1[3] = (int)w[3];
    g1[4] = (int)w[4]; g1[5] = (int)w[5]; g1[6] = (int)w[6]; g1[7] = (int)w[7];

    i32x4 g2;  // tensor_dim2=1, tensor_dim3=1, dim2_stride=0, tile_dim3=0
    g2[0] = 1; g2[1] = 1; g2[2] = 0; g2[3] = 0;
    i32x4 g3;  // dim3_stride=0, tensor_dim4=1 (bit 48), tile_dim4=0
    g3[0] = 0; g3[1] = 0x10000; g3[2] = 0; g3[3] = 0;

    i32x8 gx;  // extension group (6-arg form): zero-filled
    gx[0] = 0; gx[1] = 0; gx[2] = 0; gx[3] = 0;
    gx[4] = 0; gx[5] = 0; gx[6] = 0; gx[7] = 0;

    __builtin_amdgcn_tensor_load_to_lds(g0, g1, g2, g3, gx, 0);
}
#endif

// ---------------------------------------------------------------------------
// Pass 1: per-(b,c,chunk) local EMA scan starting from 0; emit chunk carry.
// ---------------------------------------------------------------------------
__global__ void __launch_bounds__(288)
carry_kernel(const float* __restrict__ inp, const float* __restrict__ alpha,
             float* __restrict__ carry) {
    const int bc = blockIdx.y;           // b*C + c
    const int c  = bc % C_;
    const int k  = blockIdx.x;           // chunk index
    const int t0 = k * CHUNK;
    const int f  = threadIdx.x;

#if USE_TDM
    __shared__ float tile[CHUNK * ROWF]; // 25 * 514 * 4B = 51400 B
    if (threadIdx.x < 32) {              // wave 0 issues the DMA (EXEC ignored)
        const float* src = inp + (size_t)(bc * T_ + t0) * ROWF;
        tdm_load_rows(src, (u32)(uintptr_t)tile, CHUNK);
        __builtin_amdgcn_s_wait_tensorcnt(0);
    }
    __syncthreads();
#endif
    if (f >= F_) return;

    const float a = sigmoidf_(alpha[c * F_ + f]);
    const float d = 1.0f - a;
    float s = 0.0f;
#if USE_TDM
#pragma unroll 5
    for (int t = 0; t < CHUNK; ++t) {
        const float x0 = tile[t * ROWF + 2 * f];
        const float x1 = tile[t * ROWF + 2 * f + 1];
        s = fmaf(s, d, (x0 * x0 + x1 * x1) * a);
    }
#else
    const float2* row = (const float2*)inp + (size_t)(bc * T_ + t0) * F_ + f;
#pragma unroll 5
    for (int t = 0; t < CHUNK; ++t) {
        const float2 v = row[(size_t)t * F_];
        s = fmaf(s, d, (v.x * v.x + v.y * v.y) * a);
    }
#endif
    carry[((size_t)bc * NCHUNK + k) * F_ + f] = s;
}

// ---------------------------------------------------------------------------
// Pass 2: serial composition of the 40 chunk carries per (b,c,f).
// S_{k+1} = d^CHUNK * S_k + carry_k ; emits per-chunk start states + s_last.
// ---------------------------------------------------------------------------
__global__ void __launch_bounds__(256)
prefix_kernel(const float* __restrict__ s1, const float* __restrict__ alpha,
              const float* __restrict__ carry, float* __restrict__ start,
              float* __restrict__ s_last_out) {
    const int idx = blockIdx.x * blockDim.x + threadIdx.x; // over B*C*F
    if (idx >= B_ * C_ * F_) return;
    const int f  = idx % F_;
    const int bc = idx / F_;
    const int c  = bc % C_;

    const float a = sigmoidf_(alpha[c * F_ + f]);
    const float d = 1.0f - a;
    float factor = 1.0f;
#pragma unroll
    for (int i = 0; i < CHUNK; ++i) factor *= d;   // d^CHUNK

    float S = s1[idx];                             // initial state s_1[b,c,f]
#pragma unroll 4
    for (int k = 0; k < NCHUNK; ++k) {
        const size_t o = ((size_t)bc * NCHUNK + k) * F_ + f;
        start[o] = S;
        S = fmaf(S, factor, carry[o]);
    }
    s_last_out[idx] = S;                           // s_last output
}

// ---------------------------------------------------------------------------
// Pass 3: redo each chunk's scan seeded with its exact start state; normalize.
// Second input read should hit the 192MB L2 (input is only 66MB).
// ---------------------------------------------------------------------------
__global__ void __launch_bounds__(288)
output_kernel(const float* __restrict__ inp, const float* __restrict__ weights,
              const float* __restrict__ bias, const float* __restrict__ alpha,
              const float* __restrict__ start, float* __restrict__ out) {
    const int bc = blockIdx.y;
    const int c  = bc % C_;
    const int k  = blockIdx.x;
    const int t0 = k * CHUNK;
    const int f  = threadIdx.x;

#if USE_TDM
    __shared__ float tile[CHUNK * ROWF];
    if (threadIdx.x < 32) {
        const float* src = inp + (size_t)(bc * T_ + t0) * ROWF;
        tdm_load_rows(src, (u32)(uintptr_t)tile, CHUNK);
        __builtin_amdgcn_s_wait_tensorcnt(0);
    }
    __syncthreads();
#endif
    if (f >= F_) return;

    const float a  = sigmoidf_(alpha[c * F_ + f]);
    const float d  = 1.0f - a;
    const float wv = weights[c * F_ + f];
    const float bv = bias[c * F_ + f];
    float s = start[((size_t)bc * NCHUNK + k) * F_ + f];

    float2* orow = (float2*)out + (size_t)(bc * T_ + t0) * F_ + f;
#if USE_TDM
#pragma unroll 5
    for (int t = 0; t < CHUNK; ++t) {
        const float x0 = tile[t * ROWF + 2 * f];
        const float x1 = tile[t * ROWF + 2 * f + 1];
        s = fmaf(s, d, (x0 * x0 + x1 * x1) * a);
        const float g = wv / (sqrtf(s) + 1e-8f);
        float2 r;
        r.x = fmaf(x0, g, bv);
        r.y = fmaf(x1, g, bv);
        orow[(size_t)t * F_] = r;
    }
#else
    const float2* irow = (const float2*)inp + (size_t)(bc * T_ + t0) * F_ + f;
#pragma unroll 5
    for (int t = 0; t < CHUNK; ++t) {
        const float2 v = irow[(size_t)t * F_];
        s = fmaf(s, d, (v.x * v.x + v.y * v.y) * a);
        const float g = wv / (sqrtf(s) + 1e-8f);
        float2 r;
        r.x = fmaf(v.x, g, bv);
        r.y = fmaf(v.y, g, bv);
        orow[(size_t)t * F_] = r;
    }
#endif
}

extern "C" void kernel_launch(void* const* d_in, const int* in_sizes, int n_in,
                              void* d_out, int out_size, void* d_ws, size_t ws_size,
                              hipStream_t stream) {
    const float* inp = (const float*)d_in[0];  // [B,C,T,F,2]
    const float* wts = (const float*)d_in[1];  // [1,C,1,F,1]
    const float* bia = (const float*)d_in[2];  // [1,C,1,F,1]
    const float* alp = (const float*)d_in[3];  // [1,C,F,1]
    const float* s1  = (const float*)d_in[4];  // [B,C,F,1]

    float* out    = (float*)d_out;                          // res, then s_last
    float* s_last = out + (size_t)B_ * C_ * T_ * F_ * 2;

    // Workspace: carries + start states, each B*C*NCHUNK*F floats (~1.3 MB each)
    float* carry = (float*)d_ws;
    float* start = carry + (size_t)B_ * C_ * NCHUNK * F_;

    dim3 grid13(NCHUNK, B_ * C_);  // 40 x 32 = 1280 blocks
    carry_kernel<<<grid13, 288, 0, stream>>>(inp, alp, carry);

    const int n2 = B_ * C_ * F_;   // 8224
    prefix_kernel<<<(n2 + 255) / 256, 256, 0, stream>>>(s1, alp, carry, start, s_last);

    output_kernel<<<grid13, 288, 0, stream>>>(inp, wts, bia, alp, start, out);
}